// CGGRLoss_84181359002144
// MI455X (gfx1250) — compile-verified
//
#include <hip/hip_runtime.h>
#include <math.h>

// ---------------------------------------------------------------------------
// CGGR curriculum loss for MI455X (gfx1250, wave32).
// Pass 1: streaming online-softmax over logits (memory-bound, ~45us roofline),
//         staged through LDS with CDNA5 async global->LDS loads when available.
// Pass 2: single-workgroup finalize (reductions + 8K bitonic sort in LDS).
// ---------------------------------------------------------------------------

#define TPB_ROW 256
#define MAX_N   8192

#if defined(__has_builtin)
#if __has_builtin(__builtin_amdgcn_global_load_async_to_lds_b128) && \
    __has_builtin(__builtin_amdgcn_s_wait_asynccnt)
#define USE_ASYNC_LDS 1
#endif
#endif
#ifndef USE_ASYNC_LDS
#define USE_ASYNC_LDS 0
#endif

#if USE_ASYNC_LDS
// Builtin signature (from hipcc diagnostic): param0 is `int4 addrspace(1)*`
// (printed as `int __vector(4) __device__ *` in HIP mode).
typedef int v4i __attribute__((ext_vector_type(4)));
typedef __attribute__((address_space(1))) v4i as1_v4i;
typedef __attribute__((address_space(3))) v4i as3_v4i;
#endif

// Online-softmax accumulate of one element: running max m, s = sum exp(x-m),
// t = sum exp(x-m)*x, plus top-2 raw logits a1 >= a2.
__device__ __forceinline__ void acc_elem(float x, float& m, float& s, float& t,
                                         float& a1, float& a2) {
  if (x > m) {  // rare after warm-up for random data
    float r = __expf(m - x);
    s *= r; t *= r; m = x;
  }
  float e = __expf(x - m);
  s += e;
  t = fmaf(e, x, t);
  if (x > a1) { a2 = a1; a1 = x; }
  else if (x > a2) { a2 = x; }
}

__device__ __forceinline__ void comb_ms(float& m, float& s, float& t,
                                        float m2, float s2, float t2) {
  float nm = fmaxf(m, m2);
  float r1 = __expf(m - nm), r2 = __expf(m2 - nm);
  s = s * r1 + s2 * r2;
  t = t * r1 + t2 * r2;
  m = nm;
}

__device__ __forceinline__ void comb_top2(float& a1, float& a2, float b1, float b2) {
  // loser's own #2 is dominated by loser's #1, so this is exact top-2 merge
  float n1 = fmaxf(a1, b1);
  a2 = fmaxf(fminf(a1, b1), fmaxf(a2, b2));
  a1 = n1;
}

// ---------------------------------------------------------------------------
// Kernel 1: one block per token row. Produces ce, entropy, confidence, margin.
// ---------------------------------------------------------------------------
__global__ void __launch_bounds__(TPB_ROW)
row_stats_kernel(const float* __restrict__ logits, const int* __restrict__ targets,
                 float* __restrict__ ce_o, float* __restrict__ ent_o,
                 float* __restrict__ conf_o, float* __restrict__ marg_o, int V) {
  const int row = blockIdx.x;
  const int tid = threadIdx.x;
  const float* rowp = logits + (size_t)row * (size_t)V;
  const float4* rowp4 = (const float4*)rowp;
  const int V4 = V >> 2;                       // V is a multiple of 4 (32000)
  const int nch = (V4 + TPB_ROW - 1) / TPB_ROW;

  const float NEG_INF = -__builtin_inff();
  float m = NEG_INF, s = 0.f, t = 0.f, a1 = NEG_INF, a2 = NEG_INF;

#if USE_ASYNC_LDS
  // Double-buffered async global->LDS staging (CDNA5 ASYNCcnt pipe).
  // Addresses are clamped so every lane issues every chunk -> wave-uniform
  // ASYNCcnt bookkeeping. Each lane reads back only its own 16B, so no
  // cross-wave barrier is needed.
  __shared__ float4 stage[2][TPB_ROW];
  {
    int cl = tid < V4 ? tid : V4 - 1;
    __builtin_amdgcn_global_load_async_to_lds_b128(
        (as1_v4i*)(rowp4 + cl), (as3_v4i*)&stage[0][tid], 0, 0);
  }
  for (int c = 0; c < nch; ++c) {
    if (c + 1 < nch) {
      int nidx = (c + 1) * TPB_ROW + tid;
      int cl = nidx < V4 ? nidx : V4 - 1;
      __builtin_amdgcn_global_load_async_to_lds_b128(
          (as1_v4i*)(rowp4 + cl), (as3_v4i*)&stage[(c + 1) & 1][tid], 0, 0);
      __builtin_amdgcn_s_wait_asynccnt(1);   // chunk c has landed in LDS
    } else {
      __builtin_amdgcn_s_wait_asynccnt(0);
    }
    float4 v = stage[c & 1][tid];
    int idx4 = c * TPB_ROW + tid;
    if (idx4 < V4) {
      acc_elem(v.x, m, s, t, a1, a2);
      acc_elem(v.y, m, s, t, a1, a2);
      acc_elem(v.z, m, s, t, a1, a2);
      acc_elem(v.w, m, s, t, a1, a2);
    }
  }
#else
  for (int c = 0; c < nch; ++c) {
    int idx4 = c * TPB_ROW + tid;
    if (idx4 + TPB_ROW < V4)
      __builtin_prefetch(rowp4 + idx4 + TPB_ROW, 0, 0);  // global_prefetch_b8
    if (idx4 < V4) {
      float4 v = rowp4[idx4];                             // global_load_b128
      acc_elem(v.x, m, s, t, a1, a2);
      acc_elem(v.y, m, s, t, a1, a2);
      acc_elem(v.z, m, s, t, a1, a2);
      acc_elem(v.w, m, s, t, a1, a2);
    }
  }
#endif

  // Intra-wave (wave32) reduction.
  for (int o = 16; o; o >>= 1) {
    float m2 = __shfl_xor(m, o, 32);
    float s2 = __shfl_xor(s, o, 32);
    float t2 = __shfl_xor(t, o, 32);
    float b1 = __shfl_xor(a1, o, 32);
    float b2 = __shfl_xor(a2, o, 32);
    comb_ms(m, s, t, m2, s2, t2);
    comb_top2(a1, a2, b1, b2);
  }

  __shared__ float red[TPB_ROW / 32][5];
  const int wave = tid >> 5, lane = tid & 31;
  if (lane == 0) {
    red[wave][0] = m; red[wave][1] = s; red[wave][2] = t;
    red[wave][3] = a1; red[wave][4] = a2;
  }
  __syncthreads();

  if (tid == 0) {
    m = red[0][0]; s = red[0][1]; t = red[0][2]; a1 = red[0][3]; a2 = red[0][4];
    for (int w = 1; w < TPB_ROW / 32; ++w) {
      comb_ms(m, s, t, red[w][0], red[w][1], red[w][2]);
      comb_top2(a1, a2, red[w][3], red[w][4]);
    }
    float L = m + __logf(s);              // logsumexp
    float xt = rowp[targets[row]];        // target logit
    float p1 = __expf(a1 - L);
    float p2 = __expf(a2 - L);
    ce_o[row]   = L - xt;                 // cross-entropy
    ent_o[row]  = L - t / s;              // entropy = L - E[x]
    conf_o[row] = p1;
    marg_o[row] = p1 - p2;
  }
}

// ---------------------------------------------------------------------------
// Kernel 2: single 1024-thread block: global stats, difficulty, bitonic sort,
// dynamic-threshold masked CE mean.
// ---------------------------------------------------------------------------
__device__ __forceinline__ float block_sum(float v, float* scratch) {
  for (int o = 16; o; o >>= 1) v += __shfl_xor(v, o, 32);
  const int lane = threadIdx.x & 31, w = threadIdx.x >> 5;
  const int nw = blockDim.x >> 5;
  __syncthreads();
  if (lane == 0) scratch[w] = v;
  __syncthreads();
  float r = 0.f;
  for (int i = 0; i < nw; ++i) r += scratch[i];
  return r;
}

__device__ __forceinline__ float block_max(float v, float* scratch) {
  for (int o = 16; o; o >>= 1) v = fmaxf(v, __shfl_xor(v, o, 32));
  const int lane = threadIdx.x & 31, w = threadIdx.x >> 5;
  const int nw = blockDim.x >> 5;
  __syncthreads();
  if (lane == 0) scratch[w] = v;
  __syncthreads();
  float r = scratch[0];
  for (int i = 1; i < nw; ++i) r = fmaxf(r, scratch[i]);
  return r;
}

__global__ void __launch_bounds__(1024)
finalize_kernel(const float* __restrict__ ce, const float* __restrict__ ent,
                const float* __restrict__ conf, const float* __restrict__ marg,
                const int* __restrict__ step_cnt, float* __restrict__ out,
                int N, float inv_logV) {
  __shared__ float sbuf[MAX_N];   // 32 KB (of 320 KB LDS): sort buffer
  __shared__ float scratch[32];
  const int tid = threadIdx.x;
  const int nt = blockDim.x;

  // Phase A: max(ce) and mean(confidence)
  float lmax = -__builtin_inff(), lsum = 0.f;
  for (int i = tid; i < N; i += nt) {
    lmax = fmaxf(lmax, ce[i]);
    lsum += conf[i];
  }
  const float max_ce = block_max(lmax, scratch);
  const float avg_conf = block_sum(lsum, scratch) / (float)N;
  const float inv_mc = 1.f / (max_ce + 1e-6f);

  // Phase B: difficulty -> LDS sort buffer
  __syncthreads();
  for (int i = tid; i < N; i += nt) {
    float d = (ent[i] * inv_logV + (1.f - marg[i]) + ce[i] * inv_mc) * (1.f / 3.f);
    sbuf[i] = d;
  }
  __syncthreads();

  // Phase C: dynamic keep-count k (step_count=1000 -> progress=1, base=0.25)
  float step = (float)step_cnt[0];
  float progress = fminf(1.f, step * (1.f / 1000.f));
  float base_ratio = 1.f - progress * (1.f - 0.25f);
  float ratio = base_ratio * (1.f + 0.5f * (0.5f - avg_conf) * 2.f);
  ratio = fminf(fmaxf(ratio, 0.01f), 1.f);
  int k = (int)rintf(ratio * (float)N);
  k = k < 1 ? 1 : (k > N ? N : k);

  // Phase D: bitonic sort ascending in LDS (N = 8192 power of two)
  for (int kk = 2; kk <= N; kk <<= 1) {
    for (int j = kk >> 1; j > 0; j >>= 1) {
      for (int i = tid; i < N; i += nt) {
        int ixj = i ^ j;
        if (ixj > i) {
          float a = sbuf[i], b = sbuf[ixj];
          bool up = ((i & kk) == 0);
          if (up ? (a > b) : (a < b)) { sbuf[i] = b; sbuf[ixj] = a; }
        }
      }
      __syncthreads();
    }
  }

  // Phase E: k-th largest as threshold
  const float thresh = sbuf[N - k];

  // Phase F: masked CE mean (recompute difficulty bitwise-identically)
  float sc = 0.f, sm = 0.f;
  for (int i = tid; i < N; i += nt) {
    float d = (ent[i] * inv_logV + (1.f - marg[i]) + ce[i] * inv_mc) * (1.f / 3.f);
    if (d >= thresh) { sm += 1.f; sc += ce[i]; }
  }
  const float tot_ce = block_sum(sc, scratch);
  const float tot_m = block_sum(sm, scratch);
  if (tid == 0) out[0] = tot_ce / fmaxf(tot_m, 1.f);
}

// ---------------------------------------------------------------------------
extern "C" void kernel_launch(void* const* d_in, const int* in_sizes, int n_in,
                              void* d_out, int out_size, void* d_ws, size_t ws_size,
                              hipStream_t stream) {
  const float* logits  = (const float*)d_in[0];
  const int*   targets = (const int*)d_in[1];
  const int*   step    = (const int*)d_in[2];

  const int N = in_sizes[1];                              // 8192 token rows
  const int V = (int)((long long)in_sizes[0] / (long long)N);  // 32000

  float* ce   = (float*)d_ws;       // 4*N floats = 128 KB workspace
  float* ent  = ce + N;
  float* conf = ent + N;
  float* marg = conf + N;

  const float inv_logV = 1.0f / logf((float)V);

  row_stats_kernel<<<N, TPB_ROW, 0, stream>>>(logits, targets, ce, ent, conf,
                                              marg, V);
  finalize_kernel<<<1, 1024, 0, stream>>>(ce, ent, conf, marg, step,
                                          (float*)d_out, N, inv_logV);
}